// ContrastiveLoss_obs_9070970929208
// MI455X (gfx1250) — compile-verified
//
#include <hip/hip_runtime.h>
#include <hip/hip_bf16.h>
#include <math.h>

// ---------------------------------------------------------------------------
// Problem constants (from reference): bs=128, P=36 patches, W=80 words, D=768
// ---------------------------------------------------------------------------
#define BS   128
#define NP   36
#define NW   80
#define DIM  768
#define INV_TAU 100.0f   // 1/0.01

typedef __attribute__((ext_vector_type(2))) float v2f;
typedef __attribute__((ext_vector_type(8))) float v8f;

// f32 WMMA: D(16x16,f32) = A(16x4,f32) * B(4x16,f32) + C
// 8-arg form: (neg_a, A, neg_b, B, c_mod, C, reuse_a, reuse_b)
__device__ __forceinline__ v8f wmma_f32_4(v2f a, v2f b, v8f c) {
    return __builtin_amdgcn_wmma_f32_16x16x4_f32(false, a, false, b, (short)0, c,
                                                 false, false);
}

// ---------------------------------------------------------------------------
// Kernel 1: wordL[(b*NW+w), e] = sum_d word[b,w,d] * L[d,e]
// M = BS*NW = 10240 rows, N = K = 768.
// Each wave computes TWO adjacent 16x16 N-tiles from ONE A fragment
// (1 A-load + 2 B-loads per 2 WMMAs).
// grid: (M/16, N/(16*16)) = (640, 3), block: 256 threads = 8 waves.
// ---------------------------------------------------------------------------
__global__ void __launch_bounds__(256)
k_wordL_gemm(const float* __restrict__ word,   // [10240, 768]
             const float* __restrict__ L,      // [768, 768]
             float* __restrict__ wordL)        // [10240, 768]
{
    const int lane = threadIdx.x & 31;
    const int wave = threadIdx.x >> 5;
    const int mBase  = blockIdx.x * 16;
    const int nBase0 = (blockIdx.y * 16 + wave * 2) * 16;   // first N-tile
    const int n16    = lane & 15;
    const int koff   = (lane >> 4) * 2;       // A/B fragment K sub-offset

    const float* aRow = word + (size_t)(mBase + n16) * DIM; // lane's A row (M=lane%16)
    const int nCol0 = nBase0 + n16;
    const int nCol1 = nCol0 + 16;

    v8f acc0 = {}, acc1 = {};
    for (int k = 0; k < DIM; k += 4) {
        v2f a = *(const v2f*)(aRow + k + koff);             // A[m, k+koff .. +1]
        const float* lr0 = L + (size_t)(k + koff) * DIM;
        const float* lr1 = lr0 + DIM;
        v2f b0, b1;
        b0.x = lr0[nCol0];  b0.y = lr1[nCol0];
        b1.x = lr0[nCol1];  b1.y = lr1[nCol1];
        acc0 = wmma_f32_4(a, b0, acc0);
        acc1 = wmma_f32_4(a, b1, acc1);
    }
    // C layout: VGPR j -> M = j + 8*(lane>=16), N = lane%16
#pragma unroll
    for (int j = 0; j < 8; ++j) {
        int m = mBase + j + 8 * (lane >> 4);
        wordL[(size_t)m * DIM + nCol0] = acc0[j];
        wordL[(size_t)m * DIM + nCol1] = acc1[j];
    }
}

// ---------------------------------------------------------------------------
// Kernel 2: one block per (b,p).  WMMA the 80x36 fine tile into LDS, then the
// fused softmax-bilinear epilogue collapses it to sim[b,p].
// Stage A: waves 0..4 each own one 16-row M-tile and sweep all 3 N-tiles
// from a single A fragment per k-step (1 A-load + 3 B-loads per 3 WMMAs).
// block: 256 threads (8 waves).  grid: (128, 128).
// ---------------------------------------------------------------------------
__global__ void __launch_bounds__(256)
k_tile_sim(const float* __restrict__ wordL,   // [BS*NW, DIM]
           const float* __restrict__ patch,   // [BS, NP, DIM]
           const float* __restrict__ Ww,      // [NW, NW]
           const float* __restrict__ Wp,      // [NP, NP]
           const float* __restrict__ Ww2,     // [NW, NW]
           const float* __restrict__ Wp2,     // [NP, NP]
           float* __restrict__ sim)           // [BS, BS]
{
    const int b = blockIdx.x;
    const int p = blockIdx.y;
    const int lane = threadIdx.x & 31;
    const int wave = threadIdx.x >> 5;

    __shared__ float F[NW][49];     // 80 x 48 (cols 36..47 are pad), stride 49
    __shared__ float wl[NP];        // word_level[b,p,q]
    __shared__ float pl[NW];        // patch_level[b,w,p]
    __shared__ float halves[2];     // s2p, p2w

    // ---- Stage A: fine tile via f32 WMMA.  Waves 0..4 -> M-tiles 0..4.
    if (wave < 5) {
        const int n16  = lane & 15;
        const int koff = (lane >> 4) * 2;
        const int mBase = wave * 16;

        const float* aRow = wordL + (size_t)(b * NW + mBase + n16) * DIM;

        // B rows for the 3 N-tiles (clamp padded columns q>=36 to 35)
        const int q0 = n16;                                   // 0..15
        const int q1 = 16 + n16;                              // 16..31
        const int q2t = 32 + n16;                             // 32..47
        const int q2 = q2t < NP ? q2t : NP - 1;
        const float* bRow0 = patch + (size_t)(p * NP + q0) * DIM;
        const float* bRow1 = patch + (size_t)(p * NP + q1) * DIM;
        const float* bRow2 = patch + (size_t)(p * NP + q2) * DIM;

        v8f acc0 = {}, acc1 = {}, acc2 = {};
        for (int k = 0; k < DIM; k += 4) {
            v2f a  = *(const v2f*)(aRow  + k + koff);
            v2f b0 = *(const v2f*)(bRow0 + k + koff);
            v2f b1 = *(const v2f*)(bRow1 + k + koff);
            v2f b2 = *(const v2f*)(bRow2 + k + koff);
            acc0 = wmma_f32_4(a, b0, acc0);
            acc1 = wmma_f32_4(a, b1, acc1);
            acc2 = wmma_f32_4(a, b2, acc2);
        }
#pragma unroll
        for (int j = 0; j < 8; ++j) {
            int m = mBase + j + 8 * (lane >> 4);
            F[m][n16]      = acc0[j];
            F[m][16 + n16] = acc1[j];
            F[m][32 + n16] = acc2[j];
        }
    }
    __syncthreads();

    // ---- Stage B1: word_level[q] = sum_w softmax_w(F[:,q]/tau)[w] * (Ww[w,:] . F[:,q])
    // wave-cooperative per column (80-wide axis split over 32 lanes)
    for (int q = wave; q < NP; q += 8) {
        float m = -INFINITY;
        for (int w = lane; w < NW; w += 32) m = fmaxf(m, F[w][q]);
#pragma unroll
        for (int off = 16; off >= 1; off >>= 1)
            m = fmaxf(m, __shfl_xor(m, off, 32));

        float Z = 0.f, acc = 0.f;
        for (int w = lane; w < NW; w += 32) {
            const float* wwRow = Ww + (size_t)w * NW;
            float h = 0.f;
            for (int v = 0; v < NW; ++v) h = fmaf(wwRow[v], F[v][q], h);
            float e = __expf((F[w][q] - m) * INV_TAU);
            Z += e;
            acc = fmaf(e, h, acc);
        }
#pragma unroll
        for (int off = 16; off >= 1; off >>= 1) {
            Z   += __shfl_xor(Z, off, 32);
            acc += __shfl_xor(acc, off, 32);
        }
        if (lane == 0) wl[q] = acc / Z;
    }

    // ---- Stage B2: patch_level[w] = sum_q softmax_q(F[w,:]/tau)[q] * (Wp[q,:] . F[w,:])
    // one thread per row (36-wide axis, serial per thread)
    if (threadIdx.x < NW) {
        const int w = threadIdx.x;
        float m = -INFINITY;
        for (int q = 0; q < NP; ++q) m = fmaxf(m, F[w][q]);
        float Z = 0.f, acc = 0.f;
        for (int q = 0; q < NP; ++q) {
            const float* wpRow = Wp + (size_t)q * NP;
            float g = 0.f;
            for (int r = 0; r < NP; ++r) g = fmaf(wpRow[r], F[w][r], g);
            float e = __expf((F[w][q] - m) * INV_TAU);
            Z += e;
            acc = fmaf(e, g, acc);
        }
        pl[w] = acc / Z;
    }
    __syncthreads();

    // ---- Stage C: collapse wl (36) -> s2p  [wave 0],  pl (80) -> p2w  [wave 1]
    if (wave == 0) {
        float m = -INFINITY;
        for (int q = lane; q < NP; q += 32) m = fmaxf(m, wl[q]);
#pragma unroll
        for (int off = 16; off >= 1; off >>= 1)
            m = fmaxf(m, __shfl_xor(m, off, 32));
        float Z = 0.f, acc = 0.f;
        for (int q = lane; q < NP; q += 32) {
            const float* row = Wp2 + (size_t)q * NP;
            float g = 0.f;
            for (int r = 0; r < NP; ++r) g = fmaf(row[r], wl[r], g);
            float e = __expf((wl[q] - m) * INV_TAU);
            Z += e;
            acc = fmaf(e, g, acc);
        }
#pragma unroll
        for (int off = 16; off >= 1; off >>= 1) {
            Z   += __shfl_xor(Z, off, 32);
            acc += __shfl_xor(acc, off, 32);
        }
        if (lane == 0) halves[0] = acc / Z;
    } else if (wave == 1) {
        float m = -INFINITY;
        for (int w = lane; w < NW; w += 32) m = fmaxf(m, pl[w]);
#pragma unroll
        for (int off = 16; off >= 1; off >>= 1)
            m = fmaxf(m, __shfl_xor(m, off, 32));
        float Z = 0.f, acc = 0.f;
        for (int w = lane; w < NW; w += 32) {
            const float* row = Ww2 + (size_t)w * NW;
            float g = 0.f;
            for (int v = 0; v < NW; ++v) g = fmaf(row[v], pl[v], g);
            float e = __expf((pl[w] - m) * INV_TAU);
            Z += e;
            acc = fmaf(e, g, acc);
        }
#pragma unroll
        for (int off = 16; off >= 1; off >>= 1) {
            Z   += __shfl_xor(Z, off, 32);
            acc += __shfl_xor(acc, off, 32);
        }
        if (lane == 0) halves[1] = acc / Z;
    }
    __syncthreads();

    if (threadIdx.x == 0)
        sim[(size_t)b * BS + p] = 0.5f * (halves[0] + halves[1]);
}

// ---------------------------------------------------------------------------
// Kernel 3: loss = ( mean_i[lse_row_i - sim_ii] + mean_i[lse_col_i - sim_ii] ) / 2
// one block, 128 threads (thread i owns row i and column i).
// ---------------------------------------------------------------------------
__global__ void __launch_bounds__(128)
k_ce_loss(const float* __restrict__ sim, float* __restrict__ out)
{
    __shared__ float red[BS];
    const int i = threadIdx.x;

    const float dii = sim[(size_t)i * BS + i];

    float m = -INFINITY;
    for (int j = 0; j < BS; ++j) m = fmaxf(m, sim[(size_t)i * BS + j]);
    float Z = 0.f;
    for (int j = 0; j < BS; ++j) Z += __expf(sim[(size_t)i * BS + j] - m);
    float l1 = (m + __logf(Z)) - dii;

    float m2 = -INFINITY;
    for (int j = 0; j < BS; ++j) m2 = fmaxf(m2, sim[(size_t)j * BS + i]);
    float Z2 = 0.f;
    for (int j = 0; j < BS; ++j) Z2 += __expf(sim[(size_t)j * BS + i] - m2);
    float l2 = (m2 + __logf(Z2)) - dii;

    red[i] = l1 + l2;
    __syncthreads();
    for (int s = 64; s > 0; s >>= 1) {
        if (i < s) red[i] += red[i + s];
        __syncthreads();
    }
    if (i == 0) out[0] = red[0] / (2.0f * (float)BS);
}

// ---------------------------------------------------------------------------
// Host-side launch
// ---------------------------------------------------------------------------
extern "C" void kernel_launch(void* const* d_in, const int* in_sizes, int n_in,
                              void* d_out, int out_size, void* d_ws, size_t ws_size,
                              hipStream_t stream)
{
    const float* patch = (const float*)d_in[0];  // [128, 36, 768]
    const float* word  = (const float*)d_in[1];  // [128, 80, 768]
    const float* L     = (const float*)d_in[2];  // [768, 768]
    const float* Ww    = (const float*)d_in[3];  // [80, 80]
    const float* Wp    = (const float*)d_in[4];  // [36, 36]
    const float* Ww2   = (const float*)d_in[5];  // [80, 80]
    const float* Wp2   = (const float*)d_in[6];  // [36, 36]
    float* out = (float*)d_out;

    // workspace layout: wordL (10240*768 f32 = 31.5 MB) then sim (128*128 f32)
    float* wordL = (float*)d_ws;
    float* sim   = (float*)((char*)d_ws + (size_t)BS * NW * DIM * sizeof(float));

    // 1) wordL = word @ L
    {
        dim3 grid(BS * NW / 16, DIM / (16 * 16));  // (640, 3)
        k_wordL_gemm<<<grid, 256, 0, stream>>>(word, L, wordL);
    }
    // 2) fused fine-tile -> sim[b,p]
    {
        dim3 grid(BS, BS);                         // (128, 128)
        k_tile_sim<<<grid, 256, 0, stream>>>(wordL, patch, Ww, Wp, Ww2, Wp2, sim);
    }
    // 3) symmetric cross-entropy -> scalar
    k_ce_loss<<<1, BS, 0, stream>>>(sim, out);
}